// NodeGCN2_3659312136456
// MI455X (gfx1250) — compile-verified
//
#include <hip/hip_runtime.h>
#include <hip/hip_bf16.h>

// ---------------------------------------------------------------------------
// 2-layer GCN for MI455X (gfx1250, wave32).
//  - Dense GEMMs via v_wmma_f32_16x16x4_f32 (fp32 matrix path, keeps fp32
//    accuracy; GEMMs are memory-bound anyway: 2 GFLOP total vs ~1 GB traffic).
//  - Edge aggregation: 16 (or 8) lanes per edge, float4 gather + 4x
//    global_atomic_add_f32 scatter. Gather/scatter tables (<=26MB) live in
//    the 192MB L2, so this phase is L2-bandwidth bound by construction.
// ---------------------------------------------------------------------------

typedef float v2f __attribute__((ext_vector_type(2)));
typedef float v8f __attribute__((ext_vector_type(8)));

#define TPB 256

// ---------------- degree / norm prep ----------------------------------------

__global__ void k_deg_init(float* __restrict__ deg, int n) {
    int i = blockIdx.x * TPB + threadIdx.x;
    if (i < n) deg[i] = 1.0f;                 // self-loop contributes 1
}

__global__ void k_deg_count(const int* __restrict__ dst, float* __restrict__ deg, int E) {
    int e = blockIdx.x * TPB + threadIdx.x;
    if (e < E) atomicAdd(&deg[dst[e]], 1.0f);
}

__global__ void k_dinv(float* __restrict__ deg, int n) {
    int i = blockIdx.x * TPB + threadIdx.x;
    if (i < n) deg[i] = rsqrtf(deg[i]);       // deg >= 1 always (self-loop)
}

__global__ void k_norm(const int* __restrict__ src, const int* __restrict__ dst,
                       const float* __restrict__ dinv, float* __restrict__ nrm, int E) {
    int e = blockIdx.x * TPB + threadIdx.x;
    if (e < E) nrm[e] = dinv[src[e]] * dinv[dst[e]];
}

// ---------------- WMMA GEMM: H[M,NOUT] = A[M,K] @ W[NOUT,K]^T ---------------
// One wave computes a 16x16 tile of H. 8 waves / block.
// A-frag (16x4 f32): lane<16 -> {A[m][k+0],A[m][k+1]}, lane>=16 -> K+2..K+3.
// B-frag (4x16 f32): B[k][n] = W[n][k]; same float2 addressing on W rows.
// C/D (16x16 f32): VGPR j = row m0+8*(lane/16)+j, col n0+lane%16.

template <int K, int NOUT>
__global__ void __launch_bounds__(TPB)
gcn_gemm_wmma(const float* __restrict__ A, const float* __restrict__ W,
              float* __restrict__ H, int M) {
    constexpr int COL_TILES = NOUT / 16;          // 4 (layer1) or 2 (layer2)
    constexpr int ROW_TILES_PER_BLK = 8 / COL_TILES;

    __shared__ float lw[K * NOUT];                // 32KB / 8KB, << 320KB per WGP
    for (int i = threadIdx.x; i < K * NOUT; i += TPB) lw[i] = W[i];
    __syncthreads();

    const int wave = threadIdx.x >> 5;
    const int lane = threadIdx.x & 31;
    const int rowTile = blockIdx.x * ROW_TILES_PER_BLK + wave / COL_TILES;
    const int m0 = rowTile * 16;
    if (m0 >= M) return;                          // wave-uniform: EXEC stays all-1 for WMMA
    const int n0 = (wave % COL_TILES) * 16;

    const int half = lane >> 4;                   // 0|1
    const int l16  = lane & 15;
    const int koff = 2 * half;
    const float* __restrict__ arow = A  + (size_t)(m0 + l16) * K + koff;
    const float* __restrict__ brow = lw + (n0 + l16) * K + koff;

    v8f acc = {0.f, 0.f, 0.f, 0.f, 0.f, 0.f, 0.f, 0.f};
#pragma unroll
    for (int k = 0; k < K; k += 4) {
        v2f a = *(const v2f*)(arow + k);
        v2f b = *(const v2f*)(brow + k);
        acc = __builtin_amdgcn_wmma_f32_16x16x4_f32(
            /*neg_a=*/false, a, /*neg_b=*/false, b,
            /*c_mod=*/(short)0, acc, /*reuse_a=*/false, /*reuse_b=*/false);
    }

    const int n = n0 + l16;
    const int mbase = m0 + 8 * half;
#pragma unroll
    for (int j = 0; j < 8; ++j)
        H[(size_t)(mbase + j) * NOUT + n] = acc[j];
}

// ---------------- aggregation ----------------------------------------------

// out[i][:] = h[i][:] * dinv[i]^2   (self-loop term; also fully initializes out)
template <int F>
__global__ void k_selfinit(const float* __restrict__ h, const float* __restrict__ dinv,
                           float* __restrict__ out, int n) {
    constexpr int Q = F / 4;
    int t = blockIdx.x * TPB + threadIdx.x;
    int i = t / Q, q = t % Q;
    if (i < n) {
        float w = dinv[i] * dinv[i];
        float4 v = *(const float4*)(h + (size_t)i * F + 4 * q);
        float4 r; r.x = v.x * w; r.y = v.y * w; r.z = v.z * w; r.w = v.w * w;
        *(float4*)(out + (size_t)i * F + 4 * q) = r;
    }
}

// out[dst[e]][:] += h[src[e]][:] * norm[e]   (F/4 lanes per edge, float4 each)
template <int F>
__global__ void k_edge_agg(const float* __restrict__ h, const int* __restrict__ src,
                           const int* __restrict__ dst, const float* __restrict__ nrm,
                           float* __restrict__ out, int E) {
    constexpr int Q = F / 4;
    int t = blockIdx.x * TPB + threadIdx.x;
    int e = t / Q, q = t % Q;
    if (e < E) {
        int s = src[e], d = dst[e];
        float w = nrm[e];
        float4 v = *(const float4*)(h + (size_t)s * F + 4 * q);
        float* o = out + (size_t)d * F + 4 * q;
        atomicAdd(o + 0, v.x * w);
        atomicAdd(o + 1, v.y * w);
        atomicAdd(o + 2, v.z * w);
        atomicAdd(o + 3, v.w * w);
    }
}

template <int F, bool RELU>
__global__ void k_bias_act(float* __restrict__ io, const float* __restrict__ b, int n) {
    int t = blockIdx.x * TPB + threadIdx.x;
    if (t < n * F) {
        float v = io[t] + b[t % F];
        if (RELU) v = fmaxf(v, 0.0f);
        io[t] = v;
    }
}

// ---------------- launcher ---------------------------------------------------

static inline int cdiv(long long a, int b) { return (int)((a + b - 1) / b); }

extern "C" void kernel_launch(void* const* d_in, const int* in_sizes, int n_in,
                              void* d_out, int out_size, void* d_ws, size_t ws_size,
                              hipStream_t stream) {
    (void)n_in; (void)out_size; (void)ws_size;
    const float* x  = (const float*)d_in[0];   // [N,128]
    const int*   ei = (const int*)  d_in[1];   // [2,E] flat
    const float* W1 = (const float*)d_in[2];   // [64,128]
    const float* b1 = (const float*)d_in[3];   // [64]
    const float* W2 = (const float*)d_in[4];   // [32,64]
    const float* b2 = (const float*)d_in[5];   // [32]
    float* out = (float*)d_out;                // [N,32]

    const int N = in_sizes[0] / 128;
    const int E = in_sizes[1] / 2;
    const int* src = ei;
    const int* dst = ei + E;

    float* ws   = (float*)d_ws;
    const size_t NA = ((size_t)N + 3) & ~(size_t)3;
    float* dinv = ws;                          // N floats
    float* h1   = ws + NA;                     // 64N (h2 reuses first 32N later)
    float* agg1 = h1 + 64 * NA;                // 64N
    float* nrm  = agg1 + 64 * NA;              // E
    float* h2   = h1;                          // h1 dead once agg1 is built

    // degree / normalization (shared by both layers)
    k_deg_init <<<cdiv(N, TPB), TPB, 0, stream>>>(dinv, N);
    k_deg_count<<<cdiv(E, TPB), TPB, 0, stream>>>(dst, dinv, E);
    k_dinv     <<<cdiv(N, TPB), TPB, 0, stream>>>(dinv, N);
    k_norm     <<<cdiv(E, TPB), TPB, 0, stream>>>(src, dst, dinv, nrm, E);

    // ---- layer 1: h1 = x @ W1^T ; agg1 = scatter(norm * h1[src]) ; relu(+b1)
    {
        int rowTiles = (N + 15) / 16;
        gcn_gemm_wmma<128, 64><<<cdiv(rowTiles, 2), TPB, 0, stream>>>(x, W1, h1, N);
    }
    k_selfinit<64><<<cdiv((long long)N * 16, TPB), TPB, 0, stream>>>(h1, dinv, agg1, N);
    k_edge_agg<64><<<cdiv((long long)E * 16, TPB), TPB, 0, stream>>>(h1, src, dst, nrm, agg1, E);
    k_bias_act<64, true><<<cdiv((long long)N * 64, TPB), TPB, 0, stream>>>(agg1, b1, N);

    // ---- layer 2: h2 = agg1 @ W2^T ; out = scatter(norm * h2[src]) + b2
    {
        int rowTiles = (N + 15) / 16;
        gcn_gemm_wmma<64, 32><<<cdiv(rowTiles, 4), TPB, 0, stream>>>(agg1, W2, h2, N);
    }
    k_selfinit<32><<<cdiv((long long)N * 8, TPB), TPB, 0, stream>>>(h2, dinv, out, N);
    k_edge_agg<32><<<cdiv((long long)E * 8, TPB), TPB, 0, stream>>>(h2, src, dst, nrm, out, E);
    k_bias_act<32, false><<<cdiv((long long)N * 32, TPB), TPB, 0, stream>>>(out, b2, N);
}